// YoloNASPoseLoss_90529320665205
// MI455X (gfx1250) — compile-verified
//
#include <hip/hip_runtime.h>
#include <hip/hip_bf16.h>
#include <stdint.h>

// ---------------------------------------------------------------------------
// YOLO-NAS pose task-aligned assigner for MI455X (gfx1250, wave32).
// Memory-bound workload (~1 FLOP/byte) -> optimize data movement:
//  - only IoU tensor materialized (53.8MB), everything else recomputed
//  - top-k membership + final mask as bitmasks (1.7MB each), built with
//    wave32 __ballot (no atomics)
//  - exact stable top-k via 13 argmax passes over an LDS-resident metric row
//  - CDNA5 paths: global_load_async_to_lds + s_wait_asynccnt (guarded),
//    global_prefetch_b8 via __builtin_prefetch
// ---------------------------------------------------------------------------

#define BDIM 256

namespace {
constexpr int   B    = 32;
constexpr int   L    = 8400;
constexpr int   C    = 1;
constexpr int   NGT  = 50;
constexpr int   KPT  = 17;
constexpr int   POSE = KPT * 3;            // 51
constexpr int   TOPK = 13;
constexpr float EPSF = 1e-9f;
constexpr int   WPL  = 264;                // ceil(8400/32)=263, padded

// flat output offsets (labels, bboxes, poses, scores, agi_flat)
constexpr size_t OL = 0;
constexpr size_t OB = (size_t)B * L;
constexpr size_t OP = OB + (size_t)B * L * 4;
constexpr size_t OS = OP + (size_t)B * L * POSE;
constexpr size_t OA = OS + (size_t)B * L * C;
} // namespace

#if defined(__has_builtin)
#if __has_builtin(__builtin_amdgcn_global_load_async_to_lds_b32) && \
    __has_builtin(__builtin_amdgcn_s_wait_asynccnt)
#define USE_ASYNC_LDS 1
#endif
#endif

#if defined(USE_ASYNC_LDS)
typedef __attribute__((address_space(1))) int* as1_i32;
typedef __attribute__((address_space(3))) int* as3_i32;
// One 32-bit async global->LDS copy (per-lane addresses, tracked by ASYNCcnt).
__device__ __forceinline__ void async_b32(const void* gsrc, void* ldst) {
  __builtin_amdgcn_global_load_async_to_lds_b32(
      (as1_i32)gsrc, (as3_i32)ldst, 0, 0);
}
#endif

__device__ __forceinline__ float pow6(float x) {
  float t = x * x;
  return t * t * t;
}

__device__ __forceinline__ float iou_fn(float4 a, float4 b) {
  float ix = fmaxf(fminf(a.z, b.z) - fmaxf(a.x, b.x), 0.f);
  float iy = fmaxf(fminf(a.w, b.w) - fmaxf(a.y, b.y), 0.f);
  float inter = ix * iy;
  float a1 = (a.z - a.x) * (a.w - a.y);
  float a2 = (b.z - b.x) * (b.w - b.y);
  return inter / (a1 + a2 - inter + EPSF);
}

__device__ __forceinline__ bool in_box(float ax, float ay,
                                       float x1, float y1, float x2, float y2) {
  float d = fminf(fminf(ax - x1, ay - y1), fminf(x2 - ax, y2 - ay));
  return d > EPSF;
}

// ---------------------------------------------------------------- kernel A
// ious[(b*NGT+j)*L + i] = IoU(gt[b,j], pred[b,i])
__global__ void k_iou(const float* __restrict__ gtb,
                      const float* __restrict__ pbb,
                      float* __restrict__ ious) {
  int row = blockIdx.y;                       // b*NGT + j
  int i   = blockIdx.x * BDIM + threadIdx.x;
  if (i >= L) return;
  int b = row / NGT;
  float4 g = *(const float4*)(gtb + (size_t)row * 4);
  float4 p = *(const float4*)(pbb + ((size_t)b * L + i) * 4);
  ious[(size_t)row * L + i] = iou_fn(g, p);
}

// ---------------------------------------------------------------- kernel B
// Exact stable top-13 of align*in_gts per (b,j) row -> bitmask.
__global__ void k_topk(const float* __restrict__ ious,
                       const float* __restrict__ scores,
                       const float* __restrict__ anchors,
                       const float* __restrict__ gtb,
                       const int* __restrict__ gtl,
                       const float* __restrict__ pad,
                       unsigned* __restrict__ topkbits) {
  __shared__ float metric[L];                 // 33.6KB of 320KB WGP LDS
  __shared__ unsigned long long skey[BDIM];
  int row = blockIdx.x;
  int b   = row / NGT;
  int tid = threadIdx.x;
  unsigned* bits = topkbits + (size_t)row * WPL;
  for (int w = tid; w < WPL; w += BDIM) bits[w] = 0u;
  if (!(pad[row] > 0.f)) return;              // masked GT: empty top-k row

  float4 g = *(const float4*)(gtb + (size_t)row * 4);
  int lbl = gtl[row];
  const float* irow = ious + (size_t)row * L;
  for (int i = tid; i < L; i += BDIM) {
    float iou = irow[i];
    float a = scores[((size_t)b * L + i) * C + lbl] * pow6(iou);
    float ax = anchors[2 * i], ay = anchors[2 * i + 1];
    metric[i] = in_box(ax, ay, g.x, g.y, g.z, g.w) ? a : 0.f;
  }
  __syncthreads();

  // 13 argmax passes; key = (value bits << 32) | (MAXINT - idx):
  // larger value wins, ties -> lowest index (== jax.lax.top_k stable order).
  for (int k = 0; k < TOPK; ++k) {
    unsigned long long best = 0ull;
    for (int i = tid; i < L; i += BDIM) {
      float v = metric[i];
      if (v >= 0.f) {                         // picked entries set to -1
        unsigned long long key =
            ((unsigned long long)__float_as_uint(v) << 32) |
            (unsigned)(0x7FFFFFFF - i);
        if (key > best) best = key;
      }
    }
    skey[tid] = best;
    __syncthreads();
    for (int s = BDIM / 2; s > 0; s >>= 1) {
      if (tid < s && skey[tid + s] > skey[tid]) skey[tid] = skey[tid + s];
      __syncthreads();
    }
    if (tid == 0) {
      int idx = 0x7FFFFFFF - (int)(skey[0] & 0xFFFFFFFFu);
      metric[idx] = -1.f;
      bits[idx >> 5] |= 1u << (idx & 31);
    }
    __syncthreads();
  }
}

// ---------------------------------------------------------------- kernel C
// Per-anchor: build 50-bit candidate mask, resolve mps>1 with argmax-IoU,
// emit final mask bitmask via wave32 ballot + gathered outputs.
__global__ void k_assign(const float* __restrict__ ious,
                         const float* __restrict__ anchors,
                         const float* __restrict__ gtb,
                         const int* __restrict__ gtl,
                         const float* __restrict__ gtp,
                         const float* __restrict__ pad,
                         const unsigned* __restrict__ topkbits,
                         const int* __restrict__ bg_ptr,
                         unsigned* __restrict__ maskbits,
                         int* __restrict__ lbl_ws,
                         float* __restrict__ out) {
  __shared__ float sgt[NGT * 4];
  __shared__ float spad[NGT];
  __shared__ int   slbl[NGT];
  int tid = threadIdx.x;
  int b   = blockIdx.y;

#if defined(USE_ASYNC_LDS)
  if (tid < NGT * 4)
    async_b32(gtb + (size_t)b * NGT * 4 + tid, &sgt[tid]);
  if (tid < NGT) {
    async_b32(pad + (size_t)b * NGT + tid, &spad[tid]);
    async_b32(gtl + (size_t)b * NGT + tid, &slbl[tid]);
  }
  __builtin_amdgcn_s_wait_asynccnt(0);
#else
  if (tid < NGT * 4) sgt[tid] = gtb[(size_t)b * NGT * 4 + tid];
  if (tid < NGT) {
    spad[tid] = pad[(size_t)b * NGT + tid];
    slbl[tid] = gtl[(size_t)b * NGT + tid];
  }
#endif
  __syncthreads();

  int  i   = blockIdx.x * BDIM + tid;
  bool act = (i < L);
  float ax = 0.f, ay = 0.f;
  if (act) { ax = anchors[2 * i]; ay = anchors[2 * i + 1]; }

  unsigned long long m0 = 0ull;
  int   cnt  = 0;
  float biou = -1.f;
  int   bj   = 0;
  const float*    irow = ious + (size_t)b * NGT * L + (act ? i : 0);
  const unsigned* tkb  = topkbits + (size_t)b * NGT * WPL + ((act ? i : 0) >> 5);

  for (int j = 0; j < NGT; ++j) {
    if (act) {
      float iou = irow[(size_t)j * L];
      if (j + 1 < NGT) __builtin_prefetch(&irow[(size_t)(j + 1) * L], 0, 1);
      bool ig = in_box(ax, ay, sgt[j * 4 + 0], sgt[j * 4 + 1],
                               sgt[j * 4 + 2], sgt[j * 4 + 3]);
      bool tk = (tkb[(size_t)j * WPL] >> (i & 31)) & 1u;
      if (ig && tk && (spad[j] > 0.f)) { m0 |= (1ull << j); ++cnt; }
      if (iou > biou) { biou = iou; bj = j; }     // first-max tie-break
    }
  }

  unsigned long long mf = (cnt > 1) ? (1ull << bj) : m0;
  int mps = (cnt > 1) ? 1 : cnt;
  int agi = (cnt > 1) ? bj : (cnt > 0 ? (__ffsll(m0) - 1) : 0);

  // final mask -> bitmask, one ballot per GT (wave32: low 32 bits used)
  for (int j = 0; j < NGT; ++j) {
    unsigned long long bal = __ballot(act && ((mf >> j) & 1ull));
    if ((tid & 31) == 0 && act)
      maskbits[((size_t)b * NGT + j) * WPL + (i >> 5)] = (unsigned)bal;
  }

  if (act) {
    size_t o   = (size_t)b * L + i;
    int    bg  = bg_ptr[0];
    int    src = b * NGT + agi;
    int    lbl = (mps > 0) ? slbl[agi] : bg;
    out[OL + o] = (float)lbl;
    for (int c = 0; c < 4; ++c) out[OB + o * 4 + c] = sgt[agi * 4 + c];
    const float* gp = gtp + (size_t)src * POSE;
    for (int c = 0; c < POSE; ++c) out[OP + o * POSE + c] = gp[c];
    out[OA + o] = (float)src;
    lbl_ws[o]   = lbl;
  }
}

// ---------------------------------------------------------------- kernel D
// Per (b,j): max(align*mask) and max(iou*mask) over anchors.
__global__ void k_rowmax(const float* __restrict__ ious,
                         const float* __restrict__ scores,
                         const int* __restrict__ gtl,
                         const unsigned* __restrict__ maskbits,
                         float* __restrict__ rmm,
                         float* __restrict__ rmi) {
  __shared__ float s1[BDIM], s2[BDIM];
  int row = blockIdx.x;
  int b   = row / NGT;
  int tid = threadIdx.x;
  int lbl = gtl[row];
  const float*    irow = ious + (size_t)row * L;
  const unsigned* mb   = maskbits + (size_t)row * WPL;
  float m1 = 0.f, m2 = 0.f;
  for (int i = tid; i < L; i += BDIM) {
    if ((mb[i >> 5] >> (i & 31)) & 1u) {
      float iou = irow[i];
      float al  = scores[((size_t)b * L + i) * C + lbl] * pow6(iou);
      m1 = fmaxf(m1, al);
      m2 = fmaxf(m2, iou);
    }
  }
  s1[tid] = m1; s2[tid] = m2;
  __syncthreads();
  for (int s = BDIM / 2; s > 0; s >>= 1) {
    if (tid < s) {
      s1[tid] = fmaxf(s1[tid], s1[tid + s]);
      s2[tid] = fmaxf(s2[tid], s2[tid + s]);
    }
    __syncthreads();
  }
  if (tid == 0) { rmm[row] = s1[0]; rmi[row] = s2[0]; }
}

// ---------------------------------------------------------------- kernel E
// assigned_scores = one_hot(label)[keep] * max_j(align*mask / (rmm+eps) * rmi)
__global__ void k_scores(const float* __restrict__ ious,
                         const float* __restrict__ scores,
                         const int* __restrict__ gtl,
                         const unsigned* __restrict__ maskbits,
                         const float* __restrict__ rmm,
                         const float* __restrict__ rmi,
                         const int* __restrict__ lbl_ws,
                         const int* __restrict__ bg_ptr,
                         float* __restrict__ out) {
  __shared__ int   slbl[NGT];
  __shared__ float snrm[NGT];
  int b = blockIdx.y, tid = threadIdx.x;
  if (tid < NGT) {
    slbl[tid] = gtl[(size_t)b * NGT + tid];
    snrm[tid] = rmi[(size_t)b * NGT + tid] / (rmm[(size_t)b * NGT + tid] + EPSF);
  }
  __syncthreads();
  int i = blockIdx.x * BDIM + tid;
  if (i >= L) return;
  float t = 0.f;
  for (int j = 0; j < NGT; ++j) {
    size_t row = (size_t)b * NGT + j;
    if ((maskbits[row * WPL + (i >> 5)] >> (i & 31)) & 1u) {
      float iou = ious[row * L + i];
      float al  = scores[((size_t)b * L + i) * C + slbl[j]] * pow6(iou);
      t = fmaxf(t, al * snrm[j]);
    }
  }
  int lbl = lbl_ws[(size_t)b * L + i];
  int bg  = bg_ptr[0];
  for (int c = 0; c < C; ++c) {
    int cls = c + (c >= bg ? 1 : 0);          // keep = classes != bg_index
    out[OS + ((size_t)b * L + i) * C + c] = (lbl == cls) ? t : 0.f;
  }
}

// ---------------------------------------------------------------- launch
extern "C" void kernel_launch(void* const* d_in, const int* in_sizes, int n_in,
                              void* d_out, int out_size, void* d_ws, size_t ws_size,
                              hipStream_t stream) {
  (void)in_sizes; (void)n_in; (void)out_size; (void)ws_size;
  const float* pred_scores = (const float*)d_in[0];
  const float* pred_bboxes = (const float*)d_in[1];
  /* d_in[2] pred_poses: unused by the reference assigner */
  const float* anchors     = (const float*)d_in[3];
  const int*   gt_labels   = (const int*)d_in[4];
  const float* gt_bboxes   = (const float*)d_in[5];
  const float* gt_poses    = (const float*)d_in[6];
  const float* pad_mask    = (const float*)d_in[7];
  const int*   bg_index    = (const int*)d_in[8];
  float*       out         = (float*)d_out;

  // workspace carve-up (~58 MB total)
  char* w = (char*)d_ws;
  float*    ious     = (float*)w;    w += (size_t)B * NGT * L * sizeof(float);
  unsigned* topkbits = (unsigned*)w; w += (size_t)B * NGT * WPL * sizeof(unsigned);
  unsigned* maskbits = (unsigned*)w; w += (size_t)B * NGT * WPL * sizeof(unsigned);
  int*      lbl_ws   = (int*)w;      w += (size_t)B * L * sizeof(int);
  float*    rmm      = (float*)w;    w += (size_t)B * NGT * sizeof(float);
  float*    rmi      = (float*)w;    w += (size_t)B * NGT * sizeof(float);

  dim3 gridRowL((L + BDIM - 1) / BDIM, B * NGT);  // per (b,gt) x anchor tiles
  dim3 gridBL((L + BDIM - 1) / BDIM, B);          // per b x anchor tiles

  k_iou<<<gridRowL, BDIM, 0, stream>>>(gt_bboxes, pred_bboxes, ious);
  k_topk<<<dim3(B * NGT), BDIM, 0, stream>>>(ious, pred_scores, anchors,
                                             gt_bboxes, gt_labels, pad_mask,
                                             topkbits);
  k_assign<<<gridBL, BDIM, 0, stream>>>(ious, anchors, gt_bboxes, gt_labels,
                                        gt_poses, pad_mask, topkbits, bg_index,
                                        maskbits, lbl_ws, out);
  k_rowmax<<<dim3(B * NGT), BDIM, 0, stream>>>(ious, pred_scores, gt_labels,
                                               maskbits, rmm, rmi);
  k_scores<<<gridBL, BDIM, 0, stream>>>(ious, pred_scores, gt_labels, maskbits,
                                        rmm, rmi, lbl_ws, bg_index, out);
}